// GINModel_25056839205564
// MI455X (gfx1250) — compile-verified
//
#include <hip/hip_runtime.h>

#define NNODES   100000
#define NEDGES   1600000
#define INCH     128
#define HID      64
#define NGRAPHS  128

typedef __attribute__((ext_vector_type(2))) float v2f;
typedef __attribute__((ext_vector_type(8))) float v8f;

// D = A(16x4,f32) * B(4x16,f32) + C(16x16,f32)  — native FP32 WMMA on gfx1250
__device__ __forceinline__ v8f wmma4(v2f a, v2f b, v8f c) {
    return __builtin_amdgcn_wmma_f32_16x16x4_f32(
        /*neg_a=*/false, a, /*neg_b=*/false, b,
        /*c_mod=*/(short)0, c, /*reuse_a=*/false, /*reuse_b=*/false);
}

// ---------------------------------------------------------------------------
// zero fill
// ---------------------------------------------------------------------------
__global__ void zero_kernel(float* __restrict__ p, size_t n) {
    size_t stride = (size_t)gridDim.x * blockDim.x;
    for (size_t i = (size_t)blockIdx.x * blockDim.x + threadIdx.x; i < n; i += stride)
        p[i] = 0.0f;
}

// ---------------------------------------------------------------------------
// agg[dst] += H[src]  over all edges. chunks = C/4 float4 chunks per edge;
// consecutive lanes take consecutive chunks of the same edge -> coalesced
// 128B gathers, consecutive-address f32 atomics (global_atomic_add_f32, no rtn)
// ---------------------------------------------------------------------------
__global__ void edge_agg_kernel(const float* __restrict__ H,
                                const int* __restrict__ src,
                                const int* __restrict__ dst,
                                float* __restrict__ AGG,
                                int n_edges, int shift /* log2(C/4) */) {
    const int  chunks = 1 << shift;
    const int  C      = chunks << 2;
    const long long total  = (long long)n_edges << shift;
    const long long stride = (long long)gridDim.x * blockDim.x;
    for (long long i = (long long)blockIdx.x * blockDim.x + threadIdx.x;
         i < total; i += stride) {
        int e = (int)(i >> shift);
        int q = (int)(i & (chunks - 1));
        int s = src[e];
        int d = dst[e];
        const float4 v = *reinterpret_cast<const float4*>(H + (size_t)s * C + (q << 2));
        float* p = AGG + (size_t)d * C + (q << 2);
        atomicAdd(p + 0, v.x);
        atomicAdd(p + 1, v.y);
        atomicAdd(p + 2, v.z);
        atomicAdd(p + 3, v.w);
    }
}

// ---------------------------------------------------------------------------
// Fused GIN layer for a 16-row node tile (one wave per block, EXEC all ones):
//   z = X + AGG                         (staged to LDS, padded stride)
//   t = relu(z @ W1 + b1)               (WMMA f32 16x16x4, staged to LDS)
//   OUT = relu(t @ W2 + b2)
// ---------------------------------------------------------------------------
template <int CIN>
__global__ void __launch_bounds__(32)
gin_mlp_kernel(const float* __restrict__ X, const float* __restrict__ AGG,
               const float* __restrict__ W1, const float* __restrict__ B1,
               const float* __restrict__ W2, const float* __restrict__ B2,
               float* __restrict__ OUT) {
    constexpr int ZS = CIN + 4;   // padded row stride (bank-conflict avoidance)
    constexpr int TS = HID + 4;
    __shared__ float zbuf[16 * ZS];
    __shared__ float tbuf[16 * TS];

    const int lane = threadIdx.x;
    const int row0 = blockIdx.x * 16;

    // stage z = X + AGG into LDS
    for (int i = lane; i < 16 * CIN; i += 32) {
        int r = i / CIN, c = i - r * CIN;
        size_t g = (size_t)(row0 + r) * CIN + c;
        zbuf[r * ZS + c] = X[g] + AGG[g];
    }
    __syncthreads();

    const int M  = lane & 15;          // A row
    const int kh = (lane >> 4) << 1;   // A/B half-K offset: 0 or 2
    const int N  = lane & 15;          // B/C/D column
    const int mh = (lane >> 4) << 3;   // C/D row base: 0 or 8

    // ---- GEMM1: z[16xCIN] @ W1[CINxHID] + b1 ----
    v8f acc[4];
#pragma unroll
    for (int nt = 0; nt < 4; ++nt) {
        float bv = B1[nt * 16 + N];
#pragma unroll
        for (int v = 0; v < 8; ++v) acc[nt][v] = bv;
    }
    for (int ks = 0; ks < CIN / 4; ++ks) {
        int kb = ks * 4 + kh;
        v2f a;
        a.x = zbuf[M * ZS + kb];
        a.y = zbuf[M * ZS + kb + 1];
#pragma unroll
        for (int nt = 0; nt < 4; ++nt) {
            int n = nt * 16 + N;
            v2f b;
            b.x = W1[(size_t)kb * HID + n];
            b.y = W1[(size_t)(kb + 1) * HID + n];
            acc[nt] = wmma4(a, b, acc[nt]);
        }
    }
    // relu -> tbuf
#pragma unroll
    for (int nt = 0; nt < 4; ++nt)
#pragma unroll
        for (int v = 0; v < 8; ++v)
            tbuf[(mh + v) * TS + nt * 16 + N] = fmaxf(acc[nt][v], 0.0f);
    __syncthreads();

    // ---- GEMM2: t[16xHID] @ W2[HIDxHID] + b2 ----
    v8f acc2[4];
#pragma unroll
    for (int nt = 0; nt < 4; ++nt) {
        float bv = B2[nt * 16 + N];
#pragma unroll
        for (int v = 0; v < 8; ++v) acc2[nt][v] = bv;
    }
#pragma unroll
    for (int ks = 0; ks < HID / 4; ++ks) {
        int kb = ks * 4 + kh;
        v2f a;
        a.x = tbuf[M * TS + kb];
        a.y = tbuf[M * TS + kb + 1];
#pragma unroll
        for (int nt = 0; nt < 4; ++nt) {
            int n = nt * 16 + N;
            v2f b;
            b.x = W2[(size_t)kb * HID + n];
            b.y = W2[(size_t)(kb + 1) * HID + n];
            acc2[nt] = wmma4(a, b, acc2[nt]);
        }
    }
#pragma unroll
    for (int nt = 0; nt < 4; ++nt)
#pragma unroll
        for (int v = 0; v < 8; ++v)
            OUT[(size_t)(row0 + mh + v) * HID + nt * 16 + N] = fmaxf(acc2[nt][v], 0.0f);
}

// ---------------------------------------------------------------------------
// JumpingKnowledge: OUT = concat(h0..h3)[16x256] @ Wjk[256x64] + bjk (no relu)
// ---------------------------------------------------------------------------
__global__ void __launch_bounds__(32)
jk_kernel(const float* __restrict__ H0, const float* __restrict__ H1,
          const float* __restrict__ H2, const float* __restrict__ H3,
          const float* __restrict__ WJK, const float* __restrict__ BJK,
          float* __restrict__ OUT) {
    constexpr int CIN = 4 * HID;       // 256
    constexpr int ZS  = CIN + 4;
    __shared__ float zbuf[16 * ZS];

    const int lane = threadIdx.x;
    const int row0 = blockIdx.x * 16;

    const float* __restrict__ srcs[4] = {H0, H1, H2, H3};
#pragma unroll
    for (int s = 0; s < 4; ++s)
        for (int i = lane; i < 16 * HID; i += 32) {
            int r = i >> 6, c = i & 63;
            zbuf[r * ZS + s * HID + c] = srcs[s][(size_t)(row0 + r) * HID + c];
        }
    __syncthreads();

    const int M  = lane & 15;
    const int kh = (lane >> 4) << 1;
    const int N  = lane & 15;
    const int mh = (lane >> 4) << 3;

    v8f acc[4];
#pragma unroll
    for (int nt = 0; nt < 4; ++nt) {
        float bv = BJK[nt * 16 + N];
#pragma unroll
        for (int v = 0; v < 8; ++v) acc[nt][v] = bv;
    }
    for (int ks = 0; ks < CIN / 4; ++ks) {
        int kb = ks * 4 + kh;
        v2f a;
        a.x = zbuf[M * ZS + kb];
        a.y = zbuf[M * ZS + kb + 1];
#pragma unroll
        for (int nt = 0; nt < 4; ++nt) {
            int n = nt * 16 + N;
            v2f b;
            b.x = WJK[(size_t)kb * HID + n];
            b.y = WJK[(size_t)(kb + 1) * HID + n];
            acc[nt] = wmma4(a, b, acc[nt]);
        }
    }
#pragma unroll
    for (int nt = 0; nt < 4; ++nt)
#pragma unroll
        for (int v = 0; v < 8; ++v)
            OUT[(size_t)(row0 + mh + v) * HID + nt * 16 + N] = acc[nt][v];
}

// ---------------------------------------------------------------------------
// pooled[batch[node]] += jk[node]   (global add pool)
// ---------------------------------------------------------------------------
__global__ void pool_kernel(const float* __restrict__ JK,
                            const int* __restrict__ batch,
                            float* __restrict__ pooled, int n_nodes) {
    const long long total  = (long long)n_nodes * HID;
    const long long stride = (long long)gridDim.x * blockDim.x;
    for (long long i = (long long)blockIdx.x * blockDim.x + threadIdx.x;
         i < total; i += stride) {
        int node = (int)(i >> 6);
        int c    = (int)(i & 63);
        atomicAdd(&pooled[(size_t)batch[node] * HID + c], JK[i]);
    }
}

// ---------------------------------------------------------------------------
// Classifier: z = pooled@Wc1+bc1; batchnorm(axis=0, biased var); relu; @Wc2+bc2
// Single block; tiny (128x64).
// ---------------------------------------------------------------------------
__global__ void __launch_bounds__(256)
classifier_kernel(const float* __restrict__ pooled,
                  const float* __restrict__ Wc1, const float* __restrict__ bc1,
                  const float* __restrict__ gamma, const float* __restrict__ beta,
                  const float* __restrict__ Wc2, const float* __restrict__ bc2,
                  float* __restrict__ out) {
    __shared__ float z[NGRAPHS * HID];
    __shared__ float mu[HID], rstd[HID];
    const int tid = threadIdx.x;

    for (int i = tid; i < NGRAPHS * HID; i += blockDim.x) {
        int r = i >> 6, c = i & 63;
        float acc = bc1[c];
        for (int k = 0; k < HID; ++k) acc += pooled[r * HID + k] * Wc1[k * HID + c];
        z[i] = acc;
    }
    __syncthreads();

    if (tid < HID) {
        float s = 0.0f, ss = 0.0f;
        for (int r = 0; r < NGRAPHS; ++r) {
            float v = z[r * HID + tid];
            s += v; ss += v * v;
        }
        float m = s * (1.0f / NGRAPHS);
        float var = ss * (1.0f / NGRAPHS) - m * m;
        mu[tid]   = m;
        rstd[tid] = rsqrtf(var + 1e-5f);
    }
    __syncthreads();

    for (int i = tid; i < NGRAPHS * 10; i += blockDim.x) {
        int r = i / 10, o = i - r * 10;
        float acc = bc2[o];
        for (int k = 0; k < HID; ++k) {
            float v = gamma[k] * (z[r * HID + k] - mu[k]) * rstd[k] + beta[k];
            v = fmaxf(v, 0.0f);
            acc += v * Wc2[k * 10 + o];
        }
        out[i] = acc;
    }
}

// ---------------------------------------------------------------------------
extern "C" void kernel_launch(void* const* d_in, const int* in_sizes, int n_in,
                              void* d_out, int out_size, void* d_ws, size_t ws_size,
                              hipStream_t stream) {
    const float* x     = (const float*)d_in[0];
    const int*   ei    = (const int*)d_in[1];
    const int*   src   = ei;             // edge_index[0]
    const int*   dst   = ei + NEDGES;    // edge_index[1]
    const int*   batch = (const int*)d_in[2];
    const float* W1f   = (const float*)d_in[3];
    const float* b1f   = (const float*)d_in[4];
    const float* W2f   = (const float*)d_in[5];
    const float* b2f   = (const float*)d_in[6];
    const float* W1r   = (const float*)d_in[7];
    const float* b1r   = (const float*)d_in[8];
    const float* W2r   = (const float*)d_in[9];
    const float* b2r   = (const float*)d_in[10];
    const float* Wjk   = (const float*)d_in[11];
    const float* bjk   = (const float*)d_in[12];
    const float* Wc1   = (const float*)d_in[13];
    const float* bc1   = (const float*)d_in[14];
    const float* gam   = (const float*)d_in[15];
    const float* bet   = (const float*)d_in[16];
    const float* Wc2   = (const float*)d_in[17];
    const float* bc2   = (const float*)d_in[18];

    // workspace layout (floats): h0..h3 | agg(128ch, reused as jk out) | pooled
    float* ws = (float*)d_ws;
    size_t off = 0;
    float* h[4];
    for (int i = 0; i < 4; ++i) { h[i] = ws + off; off += (size_t)NNODES * HID; }
    float* agg    = ws + off; off += (size_t)NNODES * INCH;
    float* pooled = ws + off; off += (size_t)NGRAPHS * HID;
    float* jkout  = agg;  // agg is dead after last GIN layer

    const int ntiles = NNODES / 16;   // 6250, exact

    // ---- layer 0 (CIN = 128) ----
    zero_kernel<<<2048, 256, 0, stream>>>(agg, (size_t)NNODES * INCH);
    edge_agg_kernel<<<4096, 256, 0, stream>>>(x, src, dst, agg, NEDGES, 5);
    gin_mlp_kernel<INCH><<<ntiles, 32, 0, stream>>>(x, agg, W1f, b1f, W2f, b2f, h[0]);

    // ---- layers 1..3 (CIN = 64) ----
    for (int l = 0; l < 3; ++l) {
        zero_kernel<<<2048, 256, 0, stream>>>(agg, (size_t)NNODES * HID);
        edge_agg_kernel<<<4096, 256, 0, stream>>>(h[l], src, dst, agg, NEDGES, 4);
        gin_mlp_kernel<HID><<<ntiles, 32, 0, stream>>>(
            h[l], agg, W1r + (size_t)l * HID * HID, b1r + (size_t)l * HID,
            W2r + (size_t)l * HID * HID, b2r + (size_t)l * HID, h[l + 1]);
    }

    // ---- JumpingKnowledge cat + linear ----
    jk_kernel<<<ntiles, 32, 0, stream>>>(h[0], h[1], h[2], h[3], Wjk, bjk, jkout);

    // ---- global add pool ----
    zero_kernel<<<64, 256, 0, stream>>>(pooled, (size_t)NGRAPHS * HID);
    pool_kernel<<<4096, 256, 0, stream>>>(jkout, batch, pooled, NNODES);

    // ---- classifier ----
    classifier_kernel<<<1, 256, 0, stream>>>(pooled, Wc1, bc1, gam, bet, Wc2, bc2,
                                             (float*)d_out);
}